// PoincareConvTBC_56203942035751
// MI455X (gfx1250) — compile-verified
//
#include <hip/hip_runtime.h>
#include <hip/hip_bf16.h>
#include <math.h>

// ---------------- constants (from reference) ----------------
#define T_DIM   2048
#define B_DIM   16
#define CIN     512
#define COUT    512
#define KW      3
#define MROWS   (T_DIM * B_DIM)        // 32768 flattened (t,b) rows
#define KDIM    (KW * CIN)             // 1536
#define NCHUNK  (KDIM / 32)            // 48 K-chunks of 32
#define EPS_F   1e-15f
#define BETA_RATIO 0.5771624f          // beta(768,.5)/beta(256,.5)
#define MAXNORM 0.996f                 // (1 - 4e-3)/sqrt(c), c=1
#define TS      36                     // LDS tile row stride (bank-conflict-free)

typedef __attribute__((ext_vector_type(2))) float vf2;
typedef __attribute__((ext_vector_type(8))) float vf8;

static __device__ __forceinline__ float wave_reduce_sum(float v) {
    #pragma unroll
    for (int off = 16; off > 0; off >>= 1)
        v += __shfl_xor(v, off, 32);
    return v;
}

// async global->LDS (CDNA5): per-lane 16B, tracked by ASYNCcnt
static __device__ __forceinline__ void async_b128(const float* src, float* lds_dst) {
    unsigned d = (unsigned)(uintptr_t)lds_dst;       // low 32 bits = LDS offset
    asm volatile("global_load_async_to_lds_b128 %0, %1, off"
                 :: "v"(d), "v"(src) : "memory");
}

// ---------------- kernel 1: u = logmap0(x)*BETA_RATIO ; n2 = ||u||^2 -------
__global__ __launch_bounds__(256) void k_logmap(const float* __restrict__ x,
                                                float* __restrict__ u,
                                                float* __restrict__ n2) {
    const int row  = blockIdx.x * 8 + (threadIdx.x >> 5);   // one wave per row
    const int lane = threadIdx.x & 31;
    const float4* xr = (const float4*)(x + (size_t)row * CIN);
    float4* ur = (float4*)(u + (size_t)row * CIN);
    float4 v[4];
    float ss = 0.f;
    #pragma unroll
    for (int i = 0; i < 4; i++) {
        v[i] = xr[lane + 32 * i];
        ss += v[i].x * v[i].x + v[i].y * v[i].y + v[i].z * v[i].z + v[i].w * v[i].w;
    }
    ss = wave_reduce_sum(ss);
    float yn  = fmaxf(sqrtf(ss), EPS_F);
    float arg = fminf(yn, 1.f - 1e-7f);
    float at  = atanhf(arg);
    float f   = (at / yn) * BETA_RATIO;
    #pragma unroll
    for (int i = 0; i < 4; i++) {
        float4 o;
        o.x = v[i].x * f; o.y = v[i].y * f; o.z = v[i].z * f; o.w = v[i].w * f;
        ur[lane + 32 * i] = o;
    }
    if (lane == 0) {
        float un = at * BETA_RATIO;
        n2[row] = un * un;
    }
}

// ------- kernel 2: column-normalize weight_v -> transposed w_unit ----------
// wnT is [COUT][KDIM]; also zero the boundary pad used by async A staging.
__global__ __launch_bounds__(256) void k_wnorm(const float* __restrict__ wv,
                                               float* __restrict__ wnT,
                                               float* __restrict__ zpad) {
    const int o = blockIdx.x * 256 + threadIdx.x;   // 0..511, coalesced columns
    float ss = 0.f;
    for (int i = 0; i < KDIM; i++) {
        float v = wv[(size_t)i * COUT + o];
        ss += v * v;
    }
    float inv = 1.f / fmaxf(sqrtf(ss), EPS_F);
    for (int i = 0; i < KDIM; i++)
        wnT[(size_t)o * KDIM + i] = wv[(size_t)i * COUT + o] * inv;
    if (blockIdx.x == 0 && threadIdx.x < 32) zpad[threadIdx.x] = 0.f;
}

// ---------------- kernel 3: conv-GEMM via V_WMMA_F32_16X16X4_F32 ------------
// block tile 64(M) x 128(N); 8 waves = 4(M) x 2(N); wave tile 16 x 64.
// Double-buffered LDS filled by GLOBAL_LOAD_ASYNC_TO_LDS_B128 (6 issues/chunk).
static __device__ __forceinline__ void issue_chunk(const float* __restrict__ U,
                                                   const float* __restrict__ WT,
                                                   const float* __restrict__ zpad,
                                                   int m0, int n0, int tid, int kc,
                                                   float* Asb, float* Bsb) {
    const int ks       = kc >> 9;            // conv tap 0..2
    const int kloc     = kc & (CIN - 1);
    const int rowShift = (ks - 1) * B_DIM;
    // A: 64 rows x 32 K  (512 float4 slots, 2 per thread)
    #pragma unroll
    for (int i = 0; i < 2; i++) {
        int j   = tid + i * 256;
        int row = j >> 3;
        int c4  = (j & 7) << 2;
        int gr  = m0 + row + rowShift;
        const float* src = (gr >= 0 && gr < MROWS)
                         ? (U + (size_t)gr * CIN + kloc + c4)
                         : (zpad + c4);                 // zero pad, uniform issue count
        async_b128(src, Asb + row * TS + c4);
    }
    // B (transposed): 128 cols x 32 K  (1024 float4 slots, 4 per thread)
    #pragma unroll
    for (int i = 0; i < 4; i++) {
        int j   = tid + i * 256;
        int col = j >> 3;
        int k4  = (j & 7) << 2;
        const float* src = WT + (size_t)(n0 + col) * KDIM + kc + k4;
        async_b128(src, Bsb + col * TS + k4);
    }
}

__global__ __launch_bounds__(256) void k_gemm(const float* __restrict__ U,
                                              const float* __restrict__ WT,
                                              const float* __restrict__ zpad,
                                              const float* __restrict__ n2,
                                              const float* __restrict__ wg,
                                              const float* __restrict__ bias,
                                              float* __restrict__ Y) {
    const int m0   = blockIdx.x * 64;
    const int n0   = blockIdx.y * 128;
    const int tid  = threadIdx.x;
    const int lane = tid & 31;
    const int wave = tid >> 5;
    const int wm   = wave & 3;     // M sub-tile (16 rows each)
    const int wn   = wave >> 2;    // N sub-tile (64 cols each)

    __shared__ float As[2][64 * TS];    // 2 x 9216 B
    __shared__ float Bs[2][128 * TS];   // 2 x 18432 B  (transposed: [col][k])

    vf8 acc[4];
    #pragma unroll
    for (int i = 0; i < 4; i++)
        acc[i] = (vf8){0.f,0.f,0.f,0.f,0.f,0.f,0.f,0.f};

    const int am   = wm * 16 + (lane & 15);
    const int koff = (lane >> 4) << 1;           // 0 or 2 (K-split across half-waves)
    const int bcol = wn * 64 + (lane & 15);

    issue_chunk(U, WT, zpad, m0, n0, tid, 0, As[0], Bs[0]);

    for (int c = 0; c < NCHUNK; c++) {
        const int buf = c & 1;
        if (c + 1 < NCHUNK) {
            issue_chunk(U, WT, zpad, m0, n0, tid, (c + 1) * 32, As[buf ^ 1], Bs[buf ^ 1]);
            asm volatile("s_wait_asynccnt 0x6" ::: "memory");  // chunk c landed
        } else {
            asm volatile("s_wait_asynccnt 0x0" ::: "memory");
        }
        __syncthreads();

        const float* Asb = As[buf];
        const float* Bsb = Bs[buf];
        #pragma unroll
        for (int kk = 0; kk < 32; kk += 4) {
            vf2 a = *(const vf2*)(Asb + am * TS + kk + koff);
            #pragma unroll
            for (int n = 0; n < 4; n++) {
                vf2 b = *(const vf2*)(Bsb + (bcol + n * 16) * TS + kk + koff);
                acc[n] = __builtin_amdgcn_wmma_f32_16x16x4_f32(
                    false, a, false, b, (short)0, acc[n], false, false);
            }
        }
        __syncthreads();
    }

    // ---- epilogue: s = tanh(un)/un, cx2 = tanh(un)^2; Poincare MLR math ----
    const int mbase = m0 + wm * 16 + ((lane >= 16) ? 8 : 0);
    float sArr[8], c1Arr[8], idArr[8];
    #pragma unroll
    for (int v = 0; v < 8; v++) {
        int r = mbase + v;
        float un2 = n2[r];
        if (r >= B_DIM)          un2 += n2[r - B_DIM];
        if (r < MROWS - B_DIM)   un2 += n2[r + B_DIM];
        float un  = fmaxf(sqrtf(un2), EPS_F);
        float th  = tanhf(un);
        sArr[v]  = th / un;
        float cx2 = th * th;
        c1Arr[v] = 1.f + cx2;
        idArr[v] = 1.f / fmaxf(1.f - cx2, EPS_F);
    }
    #pragma unroll
    for (int n = 0; n < 4; n++) {
        int col   = n0 + wn * 64 + n * 16 + (lane & 15);
        float g2  = 2.f * wg[col];
        float dr  = 2.f * bias[col];
        float ch  = coshf(dr);
        float sh  = sinhf(dr);
        #pragma unroll
        for (int v = 0; v < 8; v++) {
            float dot = acc[n][v];
            float num = 2.f * dot * sArr[v] * ch - c1Arr[v] * sh;
            float yv  = sinhf(g2 * asinhf(num * idArr[v]));
            Y[(size_t)(mbase + v) * COUT + col] = yv;
        }
    }
}

// ---------------- kernel 4: y /= (1+sqrt(1+||y||^2)), then project ----------
__global__ __launch_bounds__(256) void k_norm(float* __restrict__ y) {
    const int row  = blockIdx.x * 8 + (threadIdx.x >> 5);
    const int lane = threadIdx.x & 31;
    float4* yr = (float4*)(y + (size_t)row * COUT);
    float4 v[4];
    float ss = 0.f;
    #pragma unroll
    for (int i = 0; i < 4; i++) {
        v[i] = yr[lane + 32 * i];
        ss += v[i].x * v[i].x + v[i].y * v[i].y + v[i].z * v[i].z + v[i].w * v[i].w;
    }
    ss = wave_reduce_sum(ss);
    float denom = 1.f + sqrtf(1.f + ss);
    float inv   = 1.f / denom;
    float nrm   = fmaxf(sqrtf(ss) * inv, EPS_F);       // ||y/denom||
    float fac   = (nrm > MAXNORM) ? (MAXNORM / nrm) : 1.f;
    float tot   = inv * fac;
    #pragma unroll
    for (int i = 0; i < 4; i++) {
        float4 o;
        o.x = v[i].x * tot; o.y = v[i].y * tot; o.z = v[i].z * tot; o.w = v[i].w * tot;
        yr[lane + 32 * i] = o;
    }
}

// ---------------- launcher --------------------------------------------------
extern "C" void kernel_launch(void* const* d_in, const int* in_sizes, int n_in,
                              void* d_out, int out_size, void* d_ws, size_t ws_size,
                              hipStream_t stream) {
    const float* x  = (const float*)d_in[0];   // [2048,16,512]
    const float* wg = (const float*)d_in[1];   // [512]
    const float* wv = (const float*)d_in[2];   // [1536,512]
    const float* bs = (const float*)d_in[3];   // [512]
    float* out = (float*)d_out;                // [2048,16,512]

    float* u    = (float*)d_ws;                // 16,777,216 floats
    float* n2   = u + (size_t)MROWS * CIN;     // 32,768 floats
    float* wnT  = n2 + MROWS;                  // 786,432 floats ([COUT][KDIM])
    float* zpad = wnT + (size_t)COUT * KDIM;   // 32 floats (zero pad)

    k_logmap<<<MROWS / 8, 256, 0, stream>>>(x, u, n2);
    k_wnorm<<<COUT / 256, 256, 0, stream>>>(wv, wnT, zpad);
    dim3 grid(MROWS / 64, COUT / 128, 1);
    k_gemm<<<grid, 256, 0, stream>>>(u, wnT, zpad, n2, wg, bs, out);
    k_norm<<<MROWS / 8, 256, 0, stream>>>(out);
}